// D3_46033459479244
// MI455X (gfx1250) — compile-verified
//
#include <hip/hip_runtime.h>
#include <math.h>

// ---------------------------------------------------------------------------
// Shapes (fixed by the reference): B=8, C=64, H=W=256; half-res 128x128.
// Workspace requirement: ~209 MB (x2 128MB, avgL 16MB, fa 32MB, fa2 32MB,
// partials/stats small).
// ---------------------------------------------------------------------------
#define B_   8
#define C_   64
#define H_   256
#define W_   256
#define HW_  (H_ * W_)
#define H2_  128
#define W2_  128
#define HW2_ (H2_ * W2_)

typedef float v2f __attribute__((ext_vector_type(2)));
typedef float v8f __attribute__((ext_vector_type(8)));

__device__ __forceinline__ float gelu_f(float v) {
    // exact GELU: 0.5*v*(1+erf(v/sqrt(2)))
    return 0.5f * v * (1.0f + erff(v * 0.70710678118654752f));
}
__device__ __forceinline__ float sigmoid_f(float v) {
    return 1.0f / (1.0f + expf(-v));
}

// ---------------------------------------------------------------------------
// K12: fused  x = x + gelu(bn(dwconv3x3(x)))   (stage 1, LDS halo tile)
//      then   x = x + gelu(bn(conv1x1_64x64(x)))  (stage 2, WMMA f32 16x16x4)
// One workgroup = one 16x16 pixel tile x all 64 channels for one batch image.
// LDS: R0 = 18x18x64 halo (20736 f, reused for WMMA D), R1 = x1 tile [c][p]
// (16384 f), W1 = 1x1 weights padded [64][65], DW = dw weights, P = bn params.
// ---------------------------------------------------------------------------
__global__ __launch_bounds__(256) void k12_dw_pw(
    const float* __restrict__ x,
    const float* __restrict__ w_s2, const float* __restrict__ s_s2,
    const float* __restrict__ b_s2,
    const float* __restrict__ w_s3, const float* __restrict__ s_s3,
    const float* __restrict__ b_s3,
    float* __restrict__ x2)
{
    extern __shared__ float lds[];
    float* R0 = lds;                  // 20736: halo, later D [cout][p]
    float* R1 = lds + 20736;          // 16384: x1 tile [c][p]
    float* W1 = lds + 37120;          // 64*65 = 4160 padded 1x1 weights
    float* DW = W1 + 4160;            // 576 depthwise weights
    float* P  = DW + 576;             // s_s2 | b_s2 | s_s3 | b_s3 (4*64)

    const int tid = threadIdx.x;
    const int b   = blockIdx.z;
    const int ty0 = blockIdx.y * 16, tx0 = blockIdx.x * 16;

    // --- phase 0: stage parameters into LDS (pad 1x1 weight rows to 65) ---
    for (int i = tid; i < 64 * 64; i += 256)
        W1[(i >> 6) * 65 + (i & 63)] = w_s3[i];
    for (int i = tid; i < 576; i += 256) DW[i] = w_s2[i];
    if (tid < 64) {
        P[tid]       = s_s2[tid];
        P[64 + tid]  = b_s2[tid];
        P[128 + tid] = s_s3[tid];
        P[192 + tid] = b_s3[tid];
    }

    // --- phase 1: halo load 18x18 per channel, zero padding ---
    for (int i = tid; i < 64 * 324; i += 256) {
        int c  = i / 324, r = i - c * 324;
        int hy = r / 18, hx = r - hy * 18;
        int gy = ty0 - 1 + hy, gx = tx0 - 1 + hx;
        float v = 0.0f;
        if (gy >= 0 && gy < H_ && gx >= 0 && gx < W_)
            v = x[((size_t)(b * 64 + c) * H_ + gy) * W_ + gx];
        R0[i] = v;
    }
    __syncthreads();

    // --- phase 2: depthwise 3x3 + bn + gelu + residual -> R1[c][p] ---
    {
        const int px = tid & 15, py = tid >> 4;
        for (int c = 0; c < 64; ++c) {
            const float* h  = &R0[c * 324 + py * 18 + px];
            const float* w9 = &DW[c * 9];
            float acc = 0.0f;
#pragma unroll
            for (int ky = 0; ky < 3; ++ky)
#pragma unroll
                for (int kx = 0; kx < 3; ++kx)
                    acc = fmaf(w9[ky * 3 + kx], h[ky * 18 + kx], acc);
            float t = fmaf(P[c], acc, P[64 + c]);
            R1[c * 256 + tid] = h[18 + 1] + gelu_f(t);  // center + gelu(bn)
        }
    }
    __syncthreads();

    // --- phase 3: 1x1 conv 64->64 via v_wmma_f32_16x16x4_f32 ---
    // D tiles: 4 cout-tiles x 16 pixel-tiles = 64; each wave handles 8.
    // A = W1 (16 cout x 4 cin), B = R1 (4 cin x 16 pixels), K-loop of 16.
    {
        const int wave = tid >> 5, lane = tid & 31;
        const int lm = lane & 15, hi = lane >> 4;
        for (int t8 = 0; t8 < 8; ++t8) {
            const int tile  = wave * 8 + t8;
            const int cout0 = (tile >> 4) * 16;
            const int p0    = (tile & 15) * 16;
            v8f acc = {0.f, 0.f, 0.f, 0.f, 0.f, 0.f, 0.f, 0.f};
#pragma unroll 4
            for (int k0 = 0; k0 < 64; k0 += 4) {
                v2f a, bb;
                // A 16x4: VGPR0 K=2*hi, VGPR1 K=2*hi+1 (ISA layout)
                a.x = W1[(cout0 + lm) * 65 + k0 + 2 * hi];
                a.y = W1[(cout0 + lm) * 65 + k0 + 2 * hi + 1];
                // B 4x16: VGPR0 rows K={0,2}, VGPR1 rows K={1,3}
                bb.x = R1[(k0 + 2 * hi    ) * 256 + p0 + lm];
                bb.y = R1[(k0 + 2 * hi + 1) * 256 + p0 + lm];
                acc = __builtin_amdgcn_wmma_f32_16x16x4_f32(
                    false, a, false, bb, (short)0, acc, false, false);
            }
            // D 16x16: VGPR v -> rows v (lanes 0-15) and v+8 (lanes 16-31)
#pragma unroll
            for (int v = 0; v < 8; ++v)
                R0[(cout0 + v + 8 * hi) * 256 + p0 + lm] = acc[v];
        }
    }
    __syncthreads();

    // --- phase 4: bn + gelu + residual, coalesced store of x2 ---
    {
        const int px = tid & 15, py = tid >> 4;
        const int gy = ty0 + py, gx = tx0 + px;
        for (int c = 0; c < 64; ++c) {
            float t = fmaf(P[128 + c], R0[c * 256 + tid], P[192 + c]);
            float o = R1[c * 256 + tid] + gelu_f(t);
            x2[((size_t)(b * 64 + c) * H_ + gy) * W_ + gx] = o;
        }
    }
}

// ---------------------------------------------------------------------------
// Kpool: 2x2 average of x2. Left channels -> avgL (gate-branch input),
// right channels -> out[:, 32:64] directly (this IS avg_r in the output).
// ---------------------------------------------------------------------------
__global__ __launch_bounds__(256) void k_pool(
    const float* __restrict__ x2, float* __restrict__ avgL,
    float* __restrict__ out)
{
    const int i = blockIdx.x * 256 + threadIdx.x;   // B*64*HW2
    const int xx = i & 127, yy = (i >> 7) & 127;
    const int c = (i >> 14) & 63, b = i >> 20;
    const float* p = &x2[((size_t)(b * 64 + c) * H_ + 2 * yy) * W_ + 2 * xx];
    const float v = 0.25f * (p[0] + p[1] + p[W_] + p[W_ + 1]);
    if (c < 32)
        avgL[((size_t)(b * 32 + c) * HW2_) + yy * W2_ + xx] = v;
    else
        out[((size_t)(b * 64 + c) * HW2_) + yy * W2_ + xx] = v;
}

// ---------------------------------------------------------------------------
// K3: fa = gelu(bn(conv1x1 32->64 (avgL)))  via WMMA f32 16x16x4, K=32.
// One workgroup = 16x16 pixel tile at 128x128 x all 64 output channels.
// ---------------------------------------------------------------------------
__global__ __launch_bounds__(256) void k3_fa_gemm(
    const float* __restrict__ avgL,
    const float* __restrict__ w_f1, const float* __restrict__ s_f1,
    const float* __restrict__ b_f1,
    float* __restrict__ fa)
{
    extern __shared__ float lds[];
    float* RB = lds;              // 8192: avgL tile [cin32][p256]
    float* RD = lds + 8192;       // 16384: D [cout64][p256]
    float* WF = lds + 24576;      // 64*33 = 2112 padded weights
    float* P  = WF + 2112;        // s_f1 | b_f1

    const int tid = threadIdx.x;
    const int b   = blockIdx.z;
    const int ty0 = blockIdx.y * 16, tx0 = blockIdx.x * 16;

    for (int i = tid; i < 64 * 32; i += 256)
        WF[(i >> 5) * 33 + (i & 31)] = w_f1[i];
    if (tid < 64) { P[tid] = s_f1[tid]; P[64 + tid] = b_f1[tid]; }
    for (int i = tid; i < 32 * 256; i += 256) {
        const int c = i >> 8, p = i & 255;
        const int py = p >> 4, px = p & 15;
        RB[i] = avgL[((size_t)(b * 32 + c) * H2_ + ty0 + py) * W2_ + tx0 + px];
    }
    __syncthreads();

    {
        const int wave = tid >> 5, lane = tid & 31;
        const int lm = lane & 15, hi = lane >> 4;
        for (int t8 = 0; t8 < 8; ++t8) {
            const int tile  = wave * 8 + t8;
            const int cout0 = (tile >> 4) * 16;
            const int p0    = (tile & 15) * 16;
            v8f acc = {0.f, 0.f, 0.f, 0.f, 0.f, 0.f, 0.f, 0.f};
#pragma unroll
            for (int k0 = 0; k0 < 32; k0 += 4) {
                v2f a, bb;
                a.x = WF[(cout0 + lm) * 33 + k0 + 2 * hi];
                a.y = WF[(cout0 + lm) * 33 + k0 + 2 * hi + 1];
                bb.x = RB[(k0 + 2 * hi    ) * 256 + p0 + lm];
                bb.y = RB[(k0 + 2 * hi + 1) * 256 + p0 + lm];
                acc = __builtin_amdgcn_wmma_f32_16x16x4_f32(
                    false, a, false, bb, (short)0, acc, false, false);
            }
#pragma unroll
            for (int v = 0; v < 8; ++v)
                RD[(cout0 + v + 8 * hi) * 256 + p0 + lm] = acc[v];
        }
    }
    __syncthreads();

    {
        const int px = tid & 15, py = tid >> 4;
        for (int c = 0; c < 64; ++c) {
            const float t = fmaf(P[c], RD[c * 256 + tid], P[64 + c]);
            fa[((size_t)(b * 64 + c) * H2_ + ty0 + py) * W2_ + tx0 + px] =
                gelu_f(t);
        }
    }
}

// ---------------------------------------------------------------------------
// K4: four depthwise branch convs (+bn+relu), residual add -> fa2.
// Also emits deterministic per-block (sum, sumsq) partials for group-norm.
// Each 256-thread block covers 2 rows of one (b,c) plane -> single group.
// ---------------------------------------------------------------------------
__global__ __launch_bounds__(256) void k4_branches(
    const float* __restrict__ fa,
    const float* __restrict__ w21, const float* __restrict__ s21,
    const float* __restrict__ b21,
    const float* __restrict__ w22, const float* __restrict__ s22,
    const float* __restrict__ b22,
    const float* __restrict__ w23, const float* __restrict__ s23,
    const float* __restrict__ b23,
    const float* __restrict__ w24, const float* __restrict__ s24,
    const float* __restrict__ b24,
    float* __restrict__ fa2, float* __restrict__ partials)
{
    __shared__ float rs[256], rq[256];
    const int i = blockIdx.x * 256 + threadIdx.x;
    const int xx = i & 127, yy = (i >> 7) & 127;
    const int c = (i >> 14) & 63, b = i >> 20;
    const int branch = c >> 4, lc = c & 15;
    const float* base = &fa[(size_t)(b * 64 + c) * HW2_];

    auto at = [&](int y2, int x2v) -> float {
        if (y2 < 0 || y2 >= H2_ || x2v < 0 || x2v >= W2_) return 0.0f;
        return base[y2 * W2_ + x2v];
    };

    float acc = 0.0f, scale, bias;
    if (branch == 0) {                       // 3x3 pad 1
        const float* w = &w21[lc * 9];
#pragma unroll
        for (int ky = 0; ky < 3; ++ky)
#pragma unroll
            for (int kx = 0; kx < 3; ++kx)
                acc = fmaf(w[ky * 3 + kx], at(yy - 1 + ky, xx - 1 + kx), acc);
        scale = s21[lc]; bias = b21[lc];
    } else if (branch == 1) {                // 3x3 dil 2, pad 2
        const float* w = &w22[lc * 9];
#pragma unroll
        for (int ky = 0; ky < 3; ++ky)
#pragma unroll
            for (int kx = 0; kx < 3; ++kx)
                acc = fmaf(w[ky * 3 + kx],
                           at(yy - 2 + 2 * ky, xx - 2 + 2 * kx), acc);
        scale = s22[lc]; bias = b22[lc];
    } else if (branch == 2) {                // 1x5, pad (0,2)
        const float* w = &w23[lc * 5];
#pragma unroll
        for (int k = 0; k < 5; ++k)
            acc = fmaf(w[k], at(yy, xx - 2 + k), acc);
        scale = s23[lc]; bias = b23[lc];
    } else {                                 // 5x1, pad (2,0)
        const float* w = &w24[lc * 5];
#pragma unroll
        for (int k = 0; k < 5; ++k)
            acc = fmaf(w[k], at(yy - 2 + k, xx), acc);
        scale = s24[lc]; bias = b24[lc];
    }
    float t = fmaf(scale, acc, bias);
    t = t > 0.0f ? t : 0.0f;                 // relu
    const float f2 = base[yy * W2_ + xx] + t;
    fa2[i] = f2;

    rs[threadIdx.x] = f2;
    rq[threadIdx.x] = f2 * f2;
    __syncthreads();
    for (int s = 128; s > 0; s >>= 1) {
        if (threadIdx.x < s) {
            rs[threadIdx.x] += rs[threadIdx.x + s];
            rq[threadIdx.x] += rq[threadIdx.x + s];
        }
        __syncthreads();
    }
    if (threadIdx.x == 0) {
        partials[2 * blockIdx.x]     = rs[0];
        partials[2 * blockIdx.x + 1] = rq[0];
    }
}

// ---------------------------------------------------------------------------
// K5: reduce K4 partials to (mean, 1/sqrt(var+eps)) per (batch, group).
// Block j = b*4 + g reduces its 1024 contiguous partial pairs (fixed order).
// ---------------------------------------------------------------------------
__global__ __launch_bounds__(256) void k5_stats(
    const float* __restrict__ partials, float* __restrict__ stats)
{
    __shared__ float rs[256], rq[256];
    const int j = blockIdx.x;                // 0..31
    const int b = j >> 2, g = j & 3;
    const int base = (b * 64 + g * 16) * (HW2_ / 256);
    float s = 0.0f, q = 0.0f;
    for (int k = threadIdx.x; k < 16 * (HW2_ / 256); k += 256) {
        s += partials[2 * (base + k)];
        q += partials[2 * (base + k) + 1];
    }
    rs[threadIdx.x] = s;
    rq[threadIdx.x] = q;
    __syncthreads();
    for (int st = 128; st > 0; st >>= 1) {
        if (threadIdx.x < st) {
            rs[threadIdx.x] += rs[threadIdx.x + st];
            rq[threadIdx.x] += rq[threadIdx.x + st];
        }
        __syncthreads();
    }
    if (threadIdx.x == 0) {
        const float N    = 16.0f * (float)HW2_;
        const float mean = rs[0] / N;
        const float var  = rq[0] / N - mean * mean;
        stats[2 * j]     = mean;
        stats[2 * j + 1] = rsqrtf(var + 1e-5f);
    }
}

// ---------------------------------------------------------------------------
// K6: per output pixel: group-norm + gelu over 64 ch, 1x1 conv 64->4 + bias,
// sigmoid -> gate; fold DCT: DCT2D entries are all +-0.5, so
//   s = u00*g~0 + u01*g~1 + u10*g~2 + u11*g~3,  out = 0.5*s  (DCT2D[0].mean).
// Writes out[:, 0:32].
// ---------------------------------------------------------------------------
__global__ __launch_bounds__(256) void k6_final(
    const float* __restrict__ fa2, const float* __restrict__ stats,
    const float* __restrict__ gn_g, const float* __restrict__ gn_b,
    const float* __restrict__ w_f3, const float* __restrict__ b_f3,
    const float* __restrict__ x2, float* __restrict__ out)
{
    __shared__ float Pg[64], Pb[64], W3[256], st[8];
    const int tid = threadIdx.x;
    const int b = blockIdx.x >> 6, tile = blockIdx.x & 63;
    if (tid < 64) { Pg[tid] = gn_g[tid]; Pb[tid] = gn_b[tid]; }
    W3[tid] = w_f3[tid];                     // (4,64) row-major
    if (tid < 8) st[tid] = stats[b * 8 + tid];
    __syncthreads();

    const int p = tile * 256 + tid;          // 0..16383
    const int yy = p >> 7, xx = p & 127;

    float g0 = 0.f, g1 = 0.f, g2 = 0.f, g3 = 0.f;
    for (int ch = 0; ch < 64; ++ch) {
        const float v   = fa2[((size_t)(b * 64 + ch) * HW2_) + p];
        const int grp   = ch >> 4;
        const float xn  = (v - st[2 * grp]) * st[2 * grp + 1];
        const float a   = gelu_f(fmaf(Pg[ch], xn, Pb[ch]));
        g0 = fmaf(W3[ch],        a, g0);
        g1 = fmaf(W3[64 + ch],   a, g1);
        g2 = fmaf(W3[128 + ch],  a, g2);
        g3 = fmaf(W3[192 + ch],  a, g3);
    }
    g0 = sigmoid_f(g0 + b_f3[0]);
    g1 = sigmoid_f(g1 + b_f3[1]);
    g2 = sigmoid_f(g2 + b_f3[2]);
    g3 = sigmoid_f(g3 + b_f3[3]);

    const float t0 = 0.5f * (g0 + g1 + g2 + g3);  // (DCT2D^T . gate)_p
    const float t1 = 0.5f * (g0 - g1 + g2 - g3);
    const float t2 = 0.5f * (g0 + g1 - g2 - g3);
    const float t3 = 0.5f * (g0 - g1 - g2 + g3);

    for (int c = 0; c < 32; ++c) {
        const float* q =
            &x2[((size_t)(b * 64 + c) * H_ + 2 * yy) * W_ + 2 * xx];
        const float s = q[0] * t0 + q[1] * t1 + q[W_] * t2 + q[W_ + 1] * t3;
        out[((size_t)(b * 64 + c) * HW2_) + p] = 0.5f * s;
    }
}

// ---------------------------------------------------------------------------
extern "C" void kernel_launch(void* const* d_in, const int* in_sizes, int n_in,
                              void* d_out, int out_size, void* d_ws,
                              size_t ws_size, hipStream_t stream)
{
    const float* x    = (const float*)d_in[0];
    const float* w_s2 = (const float*)d_in[1];
    const float* s_s2 = (const float*)d_in[2];
    const float* b_s2 = (const float*)d_in[3];
    const float* w_s3 = (const float*)d_in[4];
    const float* s_s3 = (const float*)d_in[5];
    const float* b_s3 = (const float*)d_in[6];
    const float* w_f1 = (const float*)d_in[7];
    const float* s_f1 = (const float*)d_in[8];
    const float* b_f1 = (const float*)d_in[9];
    const float* w21  = (const float*)d_in[10];
    const float* s21  = (const float*)d_in[11];
    const float* b21  = (const float*)d_in[12];
    const float* w22  = (const float*)d_in[13];
    const float* s22  = (const float*)d_in[14];
    const float* b22  = (const float*)d_in[15];
    const float* w23  = (const float*)d_in[16];
    const float* s23  = (const float*)d_in[17];
    const float* b23  = (const float*)d_in[18];
    const float* w24  = (const float*)d_in[19];
    const float* s24  = (const float*)d_in[20];
    const float* b24  = (const float*)d_in[21];
    const float* gn_g = (const float*)d_in[22];
    const float* gn_b = (const float*)d_in[23];
    const float* w_f3 = (const float*)d_in[24];
    const float* b_f3 = (const float*)d_in[25];
    float* outp = (float*)d_out;

    // workspace layout (floats)
    float* ws       = (float*)d_ws;
    float* x2       = ws;                           // 33,554,432
    float* avgL     = x2 + (size_t)B_ * C_ * HW_;   //  4,194,304
    float* fa       = avgL + (size_t)B_ * 32 * HW2_; // 8,388,608
    float* fa2      = fa + (size_t)B_ * 64 * HW2_;   // 8,388,608
    float* partials = fa2 + (size_t)B_ * 64 * HW2_;  //    65,536
    float* stats    = partials + 2 * 32768;          //        64

    const int lds12 = (20736 + 16384 + 4160 + 576 + 256) * 4;  // 168448 B
    const int lds3  = (8192 + 16384 + 2112 + 128) * 4;         // 107264 B
    hipFuncSetAttribute(reinterpret_cast<const void*>(&k12_dw_pw),
                        hipFuncAttributeMaxDynamicSharedMemorySize, lds12);
    hipFuncSetAttribute(reinterpret_cast<const void*>(&k3_fa_gemm),
                        hipFuncAttributeMaxDynamicSharedMemorySize, lds3);

    dim3 blk(256);
    k12_dw_pw<<<dim3(16, 16, 8), blk, lds12, stream>>>(
        x, w_s2, s_s2, b_s2, w_s3, s_s3, b_s3, x2);
    k_pool<<<(B_ * C_ * HW2_) / 256, blk, 0, stream>>>(x2, avgL, outp);
    k3_fa_gemm<<<dim3(8, 8, 8), blk, lds3, stream>>>(avgL, w_f1, s_f1, b_f1,
                                                     fa);
    k4_branches<<<(B_ * C_ * HW2_) / 256, blk, 0, stream>>>(
        fa, w21, s21, b21, w22, s22, b22, w23, s23, b23, w24, s24, b24, fa2,
        partials);
    k5_stats<<<32, blk, 0, stream>>>(partials, stats);
    k6_final<<<B_ * (HW2_ / 256), blk, 0, stream>>>(fa2, stats, gn_g, gn_b,
                                                    w_f3, b_f3, x2, outp);
}